// MultiScaleFusionRoIAlign_52578989637787
// MI455X (gfx1250) — compile-verified
//
#include <hip/hip_runtime.h>
#include <stdint.h>

// ---------------------------------------------------------------------------
// MultiScaleFusionRoIAlign for MI455X (gfx1250).
// Gather-bound: inputs 81MB (L2-resident in 192MB), out 38.5MB -> ~5us HBM
// floor; L2-side gather traffic dominates. Strategy:
// - Kernel 1: compose (RoIAlign bilinear o 2^l bilinear upsample) into <=4
//   taps per axis per sample, once per (roi, level). 688KB in d_ws.
// - Kernel 2: per (roi, 16-channel tile): async-stage the 896B tap table into
//   LDS (global_load_async_to_lds_b128 + s_wait_asynccnt), gather the 196
//   sample values per channel into LDS, then pool samples->7x7 bins as a
//   dense GEMM D[16ch x 49bins] = S[16ch x 196] * A[196 x 49] with
//   v_wmma_f32_16x16x4_f32 (49 K-steps, B synthesized from per-lane bitmasks).
// ---------------------------------------------------------------------------

#define OUT_HW   7
#define NSAMP    14          // OUT_HW * sampling_ratio(2)
#define NSAMP2   196         // 14*14 == 49*4 (exact K tiling)
#define N_ROIS   256
#define FUSED_H  200
#define FUSED_W  304

struct Tap { float w; int idx; };            // 8 bytes

typedef __attribute__((ext_vector_type(2))) float v2f;
typedef __attribute__((ext_vector_type(8))) float v8f;

// ---------------------------------------------------------------------------
// Kernel 1: per (roi, level, axis, sample) composed tap lists.
// Layout: taps[(((r*3+lvl)*2+axis)*14+s)*4 + k]  -> per (r,lvl) 896B contiguous.
// Zero-weight pad taps alias real indices so dead loads hit hot cache lines.
// ---------------------------------------------------------------------------
__global__ void roi_tap_kernel(const float* __restrict__ boxes,
                               Tap* __restrict__ taps) {
  int i = blockIdx.x * blockDim.x + threadIdx.x;
  const int total = N_ROIS * 3 * 2 * NSAMP;
  if (i >= total) return;
  int s   = i % NSAMP;
  int a   = (i / NSAMP) % 2;          // 0 = y axis, 1 = x axis
  int lvl = (i / (NSAMP * 2)) % 3;
  int r   = i / (NSAMP * 2 * 3);

  const int dim = (a == 0) ? FUSED_H : FUSED_W;     // fused-grid extent
  // spatial scale: 200/800 = 304/1216 = 0.25 for both axes
  float c1 = boxes[r * 4 + (a == 0 ? 1 : 0)] * 0.25f;
  float c2 = boxes[r * 4 + (a == 0 ? 3 : 2)] * 0.25f;
  float rlen = fmaxf(c2 - c1, 1.0f);
  float bsz  = rlen * (1.0f / (float)OUT_HW);
  float off  = ((float)(s & 1) + 0.5f) * 0.5f;      // 0.25 / 0.75
  float P    = c1 + ((float)(s >> 1) + off) * bsz;  // sample coord, fused grid

  bool  valid = (P >= -1.0f) && (P <= (float)dim);  // reference valid mask
  float Pc = fminf(fmaxf(P, 0.0f), (float)(dim - 1));
  int   p0 = (int)floorf(Pc);
  int   p1 = min(p0 + 1, dim - 1);
  float lf = Pc - (float)p0;
  float hf = 1.0f - lf;
  float w0 = valid ? hf : 0.0f;
  float w1 = valid ? lf : 0.0f;

  Tap t4[4];
  if (lvl == 0) {
    t4[0] = {w0, p0};
    t4[1] = {w1, p1};
    t4[2] = {0.0f, p0};                 // pad taps alias real lines
    t4[3] = {0.0f, p1};
  } else {
    // compose with half-pixel bilinear upsample (edge clamp) of coarse level
    const int   cd  = dim >> lvl;
    const float inv = (lvl == 1) ? 0.5f : 0.25f;
    int   fr[2] = {p0, p1};
    float fw[2] = {w0, w1};
    #pragma unroll
    for (int k = 0; k < 2; ++k) {
      float q = ((float)fr[k] + 0.5f) * inv - 0.5f;
      q = fminf(fmaxf(q, 0.0f), (float)(cd - 1));
      int   q0 = (int)floorf(q);
      int   q1 = min(q0 + 1, cd - 1);
      float lq = q - (float)q0;
      t4[2 * k + 0] = {fw[k] * (1.0f - lq), q0};
      t4[2 * k + 1] = {fw[k] * lq,          q1};
    }
  }
  Tap* dst = taps + (size_t)i * 4;
  #pragma unroll
  for (int k = 0; k < 4; ++k) dst[k] = t4[k];
}

// ---------------------------------------------------------------------------
// Kernel 2: gather + WMMA pooling. 128 threads = 4 wave32; EXEC stays all-1s.
// ---------------------------------------------------------------------------
__global__ __launch_bounds__(128) void roi_align_wmma_kernel(
    const float* __restrict__ f0, const float* __restrict__ f1,
    const float* __restrict__ f2, const Tap* __restrict__ taps,
    float* __restrict__ out) {
  __shared__ Tap   stap[2][NSAMP][4];     // 896 B, async-staged
  __shared__ float S[NSAMP2 * 16];        // samples x 16 channels, 12544 B

  const int t    = threadIdx.x;
  const int b    = blockIdx.x;
  const int roi  = b / 48;
  const int tile = b - roi * 48;          // 48 sixteen-channel tiles
  const int lvl  = tile >> 4;
  const int ct   = tile & 15;             // channel tile within level

  // --- async-stage this (roi,level)'s 896B tap table into LDS (async path)
  const char* gsrc =
      (const char*)(taps + (size_t)(roi * 3 + lvl) * 2 * NSAMP * 4);
  if (t < 56) {                           // 56 lanes x 16B = 896B
    uint32_t loff = (uint32_t)(uintptr_t)(&stap[0][0][0]) + (uint32_t)t * 16u;
    const void* ga = gsrc + t * 16;
    asm volatile("global_load_async_to_lds_b128 %0, %1, off"
                 :: "v"(loff), "v"(ga) : "memory");
  }
  asm volatile("s_wait_asynccnt 0" ::: "memory");
  __syncthreads();

  // --- level plane selection
  int Hc, Wc; const float* fb;
  if      (lvl == 0) { Hc = 200; Wc = 304; fb = f0; }
  else if (lvl == 1) { Hc = 100; Wc = 152; fb = f1; }
  else               { Hc = 50;  Wc = 76;  fb = f2; }
  const int plane = Hc * Wc;
  const int cch   = t & 15;                       // channel within tile
  const float* pl = fb + (size_t)(ct * 16 + cch) * plane;

  // --- phase 1: gather 196 sample values for this channel into LDS
  for (int s = (t >> 4); s < NSAMP2; s += 8) {
    int sy = s / NSAMP;
    int sx = s - sy * NSAMP;
    const Tap* ty = &stap[0][sy][0];
    const Tap* tx = &stap[1][sx][0];
    float acc = 0.0f;
    #pragma unroll
    for (int a = 0; a < 4; ++a) {
      float wy = ty[a].w;
      if (wy != 0.0f) {                           // skip zero rows (level 0)
        const float* row = pl + (size_t)ty[a].idx * Wc;
        float rs = 0.0f;
        #pragma unroll
        for (int bx = 0; bx < 4; ++bx)
          rs = fmaf(tx[bx].w, row[tx[bx].idx], rs);
        acc = fmaf(wy, rs, acc);
      }
    }
    S[s * 16 + cch] = acc;
  }
  __syncthreads();

  // --- phase 2: D[16ch x 49bins] = S[16ch x 196] * A[196 x 49], A analytic.
  // v_wmma_f32_16x16x4_f32: A 16x4 (VGPR0: K0|K2, VGPR1: K1|K3 across lane
  // halves), B 4x16 (same K split), C/D 16x16 (VGPR v: M=v | M=v+8).
  const int lane = t & 31;
  const int wv   = t >> 5;                 // 4 waves -> bins [16w, 16w+16)
  const int ncol = lane & 15;              // N index (bin column)
  const int nbin = wv * 16 + ncol;
  const int k0   = (lane & 16) ? 2 : 0;

  // Per-lane 49-bit membership masks: bit kt set iff sample (kt*4+k0[+1])
  // pools into this lane's bin. Moves all div-by-14 math out of the WMMA loop.
  uint64_t m0 = 0, m1 = 0;
  for (int kt = 0; kt < 49; ++kt) {
    int s0 = kt * 4 + k0;
    int sy0 = s0 / NSAMP, sx0 = s0 - sy0 * NSAMP;
    if (((sy0 >> 1) * 7 + (sx0 >> 1)) == nbin) m0 |= (1ull << kt);
    int s1 = s0 + 1;
    int sy1 = s1 / NSAMP, sx1 = s1 - sy1 * NSAMP;
    if (((sy1 >> 1) * 7 + (sx1 >> 1)) == nbin) m1 |= (1ull << kt);
  }

  v8f acc = {0.f, 0.f, 0.f, 0.f, 0.f, 0.f, 0.f, 0.f};
  for (int kt = 0; kt < 49; ++kt) {
    const int s0 = kt * 4 + k0;            // s0, s0+1 < 196 always
    v2f A;
    A.x = S[s0 * 16 + ncol];               // -> ds_load_2addr_b32 offset1:16
    A.y = S[s0 * 16 + 16 + ncol];
    v2f B;
    B.x = (m0 & 1) ? 0.25f : 0.0f;
    B.y = (m1 & 1) ? 0.25f : 0.0f;
    m0 >>= 1;
    m1 >>= 1;
    acc = __builtin_amdgcn_wmma_f32_16x16x4_f32(
        /*neg_a=*/false, A, /*neg_b=*/false, B,
        /*c_mod=*/(short)0, acc, /*reuse_a=*/false, /*reuse_b=*/false);
  }

  // --- store D fragment: out[roi][lvl*256 + ct*16 + ch][7][7]
  if (nbin < 49) {
    const int chhi = (lane & 16) ? 8 : 0;
    size_t base = ((size_t)roi * 768 + (size_t)(lvl * 256 + ct * 16)) * 49
                  + (size_t)nbin;
    #pragma unroll
    for (int v = 0; v < 8; ++v)
      out[base + (size_t)(v + chhi) * 49] = acc[v];
  }
}

// ---------------------------------------------------------------------------
extern "C" void kernel_launch(void* const* d_in, const int* in_sizes, int n_in,
                              void* d_out, int out_size, void* d_ws,
                              size_t ws_size, hipStream_t stream) {
  const float* f0    = (const float*)d_in[0];   // [1,256,200,304]
  const float* f1    = (const float*)d_in[1];   // [1,256,100,152]
  const float* f2    = (const float*)d_in[2];   // [1,256,50,76]
  const float* boxes = (const float*)d_in[3];   // [256,4]
  // d_in[4]/d_in[5] = img_h/img_w scalars; baked as 0.25 spatial scale.
  (void)in_sizes; (void)n_in; (void)out_size; (void)ws_size;

  Tap*   taps = (Tap*)d_ws;                     // 688,128 B of scratch
  float* out  = (float*)d_out;                  // [256,768,7,7] fp32

  const int total1 = N_ROIS * 3 * 2 * NSAMP;    // 21504 tap entries
  roi_tap_kernel<<<(total1 + 255) / 256, 256, 0, stream>>>(boxes, taps);
  roi_align_wmma_kernel<<<N_ROIS * 48, 128, 0, stream>>>(f0, f1, f2, taps, out);
}